// LML_9869834847072
// MI455X (gfx1250) — compile-verified
//
#include <hip/hip_runtime.h>
#include <hip/hip_bf16.h>
#include <math.h>

#define N_NODES 4096
#define DIM     512
#define F_OUT   64
#define HID     1024
#define ALPHA   0.2f
#define MASKV   (-9.0e15f)

typedef float v2f __attribute__((ext_vector_type(2)));
typedef float v8f __attribute__((ext_vector_type(8)));

__device__ __forceinline__ v8f wmma_f32(v2f a, v2f b, v8f c) {
    // V_WMMA_F32_16X16X4_F32 : D(16x16 f32) = A(16x4 f32) * B(4x16 f32) + C
    return __builtin_amdgcn_wmma_f32_16x16x4_f32(
        /*neg_a=*/false, a, /*neg_b=*/false, b,
        /*c_mod=*/(short)0, c, /*reuse_a=*/false, /*reuse_b=*/false);
}

// ---------------------------------------------------------------- LayerNorm
// One 256-thread block per row of 512; two elements per thread.
__global__ void LML_ln_kernel(const float* __restrict__ x,
                              const float* __restrict__ g,
                              const float* __restrict__ b,
                              float* __restrict__ out) {
    __shared__ float sbuf[256];
    const int row = blockIdx.x;
    const int tid = threadIdx.x;
    const float* xr = x + (size_t)row * DIM;
    float v0 = xr[tid];
    float v1 = xr[tid + 256];

    sbuf[tid] = v0 + v1;
    __syncthreads();
    for (int st = 128; st > 0; st >>= 1) {
        if (tid < st) sbuf[tid] += sbuf[tid + st];
        __syncthreads();
    }
    const float mean = sbuf[0] * (1.0f / DIM);
    __syncthreads();

    const float d0 = v0 - mean, d1 = v1 - mean;
    sbuf[tid] = d0 * d0 + d1 * d1;
    __syncthreads();
    for (int st = 128; st > 0; st >>= 1) {
        if (tid < st) sbuf[tid] += sbuf[tid + st];
        __syncthreads();
    }
    const float rstd = rsqrtf(sbuf[0] * (1.0f / DIM) + 1e-5f);

    float* orow = out + (size_t)row * DIM;
    orow[tid]       = d0 * rstd * g[tid]       + b[tid];
    orow[tid + 256] = d1 * rstd * g[tid + 256] + b[tid + 256];
}

// ---------------------------------------------------------------- Wh = h @ W
// A: [4096,512] row-major.  B: [512,64] row-major.  C: [4096,64].
// grid 256 (M-tiles of 16), block 128 = 4 waves, wave w -> cols [16w,16w+16)
__global__ void LML_gemm_wh(const float* __restrict__ A,
                            const float* __restrict__ B,
                            float* __restrict__ C) {
    const int wave = threadIdx.x >> 5;
    const int lane = threadIdx.x & 31;
    const int lr   = lane & 15;
    const bool hi  = lane >= 16;
    const int row0 = blockIdx.x * 16;
    const int n0   = wave * 16;
    const int r    = row0 + lr;

    const float* arow = A + (size_t)r * DIM + (hi ? 2 : 0);
    v8f acc = {};
    for (int k = 0; k < DIM; k += 4) {
        v2f av; av.x = arow[k]; av.y = arow[k + 1];
        const int kk = k + (hi ? 2 : 0);
        v2f bv; bv.x = B[(size_t)kk * F_OUT + n0 + lr];
                bv.y = B[(size_t)(kk + 1) * F_OUT + n0 + lr];
        acc = wmma_f32(av, bv, acc);
    }
    const int rbase = row0 + (hi ? 8 : 0);
#pragma unroll
    for (int v = 0; v < 8; v++)
        C[(size_t)(rbase + v) * F_OUT + n0 + lr] = acc[v];
}

// ------------------------------------------------- f_src / f_dst projections
// one wave per row; block 256 = 8 waves, grid 512
__global__ void LML_fvec_kernel(const float* __restrict__ Wh,
                                const float* __restrict__ a,
                                float* __restrict__ fsrc,
                                float* __restrict__ fdst) {
    const int row  = blockIdx.x * 8 + (threadIdx.x >> 5);
    const int lane = threadIdx.x & 31;
    const float* w = Wh + (size_t)row * F_OUT;
    float s0 = w[lane] * a[lane]      + w[lane + 32] * a[lane + 32];
    float s1 = w[lane] * a[64 + lane] + w[lane + 32] * a[96 + lane];
    for (int o = 16; o > 0; o >>= 1) {
        s0 += __shfl_down(s0, o, 32);
        s1 += __shfl_down(s1, o, 32);
    }
    if (lane == 0) { fsrc[row] = s0; fdst[row] = s1; }
}

// --------------------------------------- per-row masked softmax max and sum
// block 256 per row; e_ij recomputed on the fly (never materialized)
__global__ void LML_smstats_kernel(const int* __restrict__ adj,
                                   const float* __restrict__ fsrc,
                                   const float* __restrict__ fdst,
                                   float* __restrict__ mrow,
                                   float* __restrict__ lrow) {
    __shared__ float sbuf[256];
    const int row = blockIdx.x;
    const int tid = threadIdx.x;
    const int* arow = adj + (size_t)row * N_NODES;
    const float fs = fsrc[row];

    float mx = -3.0e38f;
    for (int j = tid; j < N_NODES; j += 256) {
        float e = fs + fdst[j];
        e = (e >= 0.0f) ? e : ALPHA * e;
        e = (arow[j] > 0) ? e : MASKV;
        mx = fmaxf(mx, e);
    }
    sbuf[tid] = mx;
    __syncthreads();
    for (int st = 128; st > 0; st >>= 1) {
        if (tid < st) sbuf[tid] = fmaxf(sbuf[tid], sbuf[tid + st]);
        __syncthreads();
    }
    mx = sbuf[0];
    __syncthreads();

    float sum = 0.0f;
    for (int j = tid; j < N_NODES; j += 256) {
        float e = fs + fdst[j];
        e = (e >= 0.0f) ? e : ALPHA * e;
        e = (arow[j] > 0) ? e : MASKV;
        sum += __expf(e - mx);
    }
    sbuf[tid] = sum;
    __syncthreads();
    for (int st = 128; st > 0; st >>= 1) {
        if (tid < st) sbuf[tid] += sbuf[tid + st];
        __syncthreads();
    }
    if (tid == 0) { mrow[row] = mx; lrow[row] = sbuf[0]; }
}

// --------------------------- h_prime = softmax(masked e) @ Wh  (flash-style)
// grid (64,4): blockIdx.x*4+wave -> 16-row tile (256 tiles),
// blockIdx.y -> K-chunk of 1024 columns; partials to Hp4[chunk].
// P values are computed per lane in the exact A-operand layout and fed to
// 4 back-to-back f32 WMMAs (cols 0-15,16-31,32-47,48-63).
__global__ void LML_attwh_kernel(const int* __restrict__ adj,
                                 const float* __restrict__ fsrc,
                                 const float* __restrict__ fdst,
                                 const float* __restrict__ mrow,
                                 const float* __restrict__ lrow,
                                 const float* __restrict__ Wh,
                                 float* __restrict__ Hp4) {
    const int wave = threadIdx.x >> 5;
    const int lane = threadIdx.x & 31;
    const int lr   = lane & 15;
    const bool hi  = lane >= 16;
    const int tile = blockIdx.x * 4 + wave;
    const int row0 = tile * 16;
    const int r    = row0 + lr;
    const int chunk = blockIdx.y;

    const float fs    = fsrc[r];
    const float m     = mrow[r];
    const float inv_l = 1.0f / lrow[r];
    const int* arow = adj + (size_t)r * N_NODES;

    v8f acc0 = {}, acc1 = {}, acc2 = {}, acc3 = {};
    const int jbeg = chunk * 1024, jend = jbeg + 1024;
    for (int j = jbeg; j < jend; j += 4) {
        const int jj = j + (hi ? 2 : 0);
        const int   a0 = arow[jj],   a1 = arow[jj + 1];
        const float d0 = fdst[jj],   d1 = fdst[jj + 1];
        float e0 = fs + d0; e0 = (e0 >= 0.0f) ? e0 : ALPHA * e0; e0 = (a0 > 0) ? e0 : MASKV;
        float e1 = fs + d1; e1 = (e1 >= 0.0f) ? e1 : ALPHA * e1; e1 = (a1 > 0) ? e1 : MASKV;
        v2f p; p.x = __expf(e0 - m) * inv_l;
               p.y = __expf(e1 - m) * inv_l;

        const float* b0p = Wh + (size_t)jj * F_OUT + lr;
        const float* b1p = b0p + F_OUT;
        v2f b0, b1, b2, b3;
        b0.x = b0p[0];  b0.y = b1p[0];
        b1.x = b0p[16]; b1.y = b1p[16];
        b2.x = b0p[32]; b2.y = b1p[32];
        b3.x = b0p[48]; b3.y = b1p[48];

        acc0 = wmma_f32(p, b0, acc0);
        acc1 = wmma_f32(p, b1, acc1);
        acc2 = wmma_f32(p, b2, acc2);
        acc3 = wmma_f32(p, b3, acc3);
    }
    float* out = Hp4 + (size_t)chunk * (N_NODES * F_OUT);
    const int rbase = row0 + (hi ? 8 : 0);
#pragma unroll
    for (int v = 0; v < 8; v++) {
        float* o = out + (size_t)(rbase + v) * F_OUT + lr;
        o[0]  = acc0[v];
        o[16] = acc1[v];
        o[32] = acc2[v];
        o[48] = acc3[v];
    }
}

// deterministic fixed-order reduce of the 4 K-chunk partials
__global__ void LML_reduce4_kernel(const float* __restrict__ p,
                                   float* __restrict__ out, int n) {
    const int i = blockIdx.x * blockDim.x + threadIdx.x;
    if (i < n) out[i] = ((p[i] + p[i + n]) + (p[i + 2 * n] + p[i + 3 * n]));
}

// ------------------------- x1 = x + h_prime @ fc_w^T + fc_b   (K = 64)
// grid (256,4), block 256 = 8 waves; wave -> one 16-col N-tile
__global__ void LML_attnout_kernel(const float* __restrict__ Hp,
                                   const float* __restrict__ fcw,
                                   const float* __restrict__ fcb,
                                   const float* __restrict__ x,
                                   float* __restrict__ x1) {
    const int wave = threadIdx.x >> 5;
    const int lane = threadIdx.x & 31;
    const int lr   = lane & 15;
    const bool hi  = lane >= 16;
    const int row0 = blockIdx.x * 16;
    const int n0   = (blockIdx.y * 8 + wave) * 16;
    const int r    = row0 + lr;

    const float* arow = Hp  + (size_t)r * F_OUT + (hi ? 2 : 0);
    const float* bcol = fcw + (size_t)(n0 + lr) * F_OUT + (hi ? 2 : 0); // B[k][n]=fcw[n][k]
    v8f acc = {};
    for (int k = 0; k < F_OUT; k += 4) {
        v2f av; av.x = arow[k]; av.y = arow[k + 1];
        v2f bv; bv.x = bcol[k]; bv.y = bcol[k + 1];
        acc = wmma_f32(av, bv, acc);
    }
    const float bias = fcb[n0 + lr];
    const int rbase = row0 + (hi ? 8 : 0);
#pragma unroll
    for (int v = 0; v < 8; v++) {
        const size_t idx = (size_t)(rbase + v) * DIM + n0 + lr;
        x1[idx] = x[idx] + acc[v] + bias;
    }
}

// ------------------------- M = gelu(h2 @ fc1_w^T + fc1_b)   (K = 512, N = 1024)
// grid (256,8), block 256 = 8 waves
__global__ void LML_fc1_kernel(const float* __restrict__ H2,
                               const float* __restrict__ w1,
                               const float* __restrict__ b1,
                               float* __restrict__ M) {
    const int wave = threadIdx.x >> 5;
    const int lane = threadIdx.x & 31;
    const int lr   = lane & 15;
    const bool hi  = lane >= 16;
    const int row0 = blockIdx.x * 16;
    const int n0   = (blockIdx.y * 8 + wave) * 16;
    const int r    = row0 + lr;

    const float* arow = H2 + (size_t)r * DIM + (hi ? 2 : 0);
    const float* bcol = w1 + (size_t)(n0 + lr) * DIM + (hi ? 2 : 0); // B[k][n]=w1[n][k]
    v8f acc = {};
    for (int k = 0; k < DIM; k += 4) {
        v2f av; av.x = arow[k]; av.y = arow[k + 1];
        v2f bv; bv.x = bcol[k]; bv.y = bcol[k + 1];
        acc = wmma_f32(av, bv, acc);
    }
    const float bias = b1[n0 + lr];
    const int rbase = row0 + (hi ? 8 : 0);
#pragma unroll
    for (int v = 0; v < 8; v++) {
        const float val = acc[v] + bias;
        const float gl  = 0.5f * val * (1.0f + erff(val * 0.70710678118654752f));
        M[(size_t)(rbase + v) * HID + n0 + lr] = gl;
    }
}

// ------------------------- out = x1 + M @ fc2_w^T + fc2_b   (K = 1024, N = 512)
// grid (256,4), block 256 = 8 waves
__global__ void LML_fc2_kernel(const float* __restrict__ M,
                               const float* __restrict__ w2,
                               const float* __restrict__ b2,
                               const float* __restrict__ x1,
                               float* __restrict__ out) {
    const int wave = threadIdx.x >> 5;
    const int lane = threadIdx.x & 31;
    const int lr   = lane & 15;
    const bool hi  = lane >= 16;
    const int row0 = blockIdx.x * 16;
    const int n0   = (blockIdx.y * 8 + wave) * 16;
    const int r    = row0 + lr;

    const float* arow = M  + (size_t)r * HID + (hi ? 2 : 0);
    const float* bcol = w2 + (size_t)(n0 + lr) * HID + (hi ? 2 : 0); // B[k][n]=w2[n][k]
    v8f acc = {};
    for (int k = 0; k < HID; k += 4) {
        v2f av; av.x = arow[k]; av.y = arow[k + 1];
        v2f bv; bv.x = bcol[k]; bv.y = bcol[k + 1];
        acc = wmma_f32(av, bv, acc);
    }
    const float bias = b2[n0 + lr];
    const int rbase = row0 + (hi ? 8 : 0);
#pragma unroll
    for (int v = 0; v < 8; v++) {
        const size_t idx = (size_t)(rbase + v) * DIM + n0 + lr;
        out[idx] = x1[idx] + acc[v] + bias;
    }
}

extern "C" void kernel_launch(void* const* d_in, const int* in_sizes, int n_in,
                              void* d_out, int out_size, void* d_ws, size_t ws_size,
                              hipStream_t stream) {
    const float* x     = (const float*)d_in[0];
    const int*   adj   = (const int*)  d_in[1];
    const float* W     = (const float*)d_in[2];
    const float* a     = (const float*)d_in[3];
    const float* fc_w  = (const float*)d_in[4];
    const float* fc_b  = (const float*)d_in[5];
    const float* ln1_g = (const float*)d_in[6];
    const float* ln1_b = (const float*)d_in[7];
    const float* ln2_g = (const float*)d_in[8];
    const float* ln2_b = (const float*)d_in[9];
    const float* fc1_w = (const float*)d_in[10];
    const float* fc1_b = (const float*)d_in[11];
    const float* fc2_w = (const float*)d_in[12];
    const float* fc2_b = (const float*)d_in[13];
    float* out = (float*)d_out;

    float* ws   = (float*)d_ws;
    float* hbuf = ws;                                  // 4096*512   (h, then h2)
    float* Wh   = hbuf + (size_t)N_NODES * DIM;        // 4096*64
    float* Hp   = Wh   + (size_t)N_NODES * F_OUT;      // 4096*64
    float* Hp4  = Hp   + (size_t)N_NODES * F_OUT;      // 4*4096*64  (K-split partials)
    float* fsrc = Hp4  + (size_t)4 * N_NODES * F_OUT;  // 4096
    float* fdst = fsrc + N_NODES;                      // 4096
    float* mrow = fdst + N_NODES;                      // 4096
    float* lrow = mrow + N_NODES;                      // 4096
    float* x1   = lrow + N_NODES;                      // 4096*512
    float* mbuf = x1   + (size_t)N_NODES * DIM;        // 4096*1024

    // --- GAT branch ---
    LML_ln_kernel<<<N_NODES, 256, 0, stream>>>(x, ln1_g, ln1_b, hbuf);
    LML_gemm_wh<<<N_NODES / 16, 128, 0, stream>>>(hbuf, W, Wh);
    LML_fvec_kernel<<<N_NODES / 8, 256, 0, stream>>>(Wh, a, fsrc, fdst);
    LML_smstats_kernel<<<N_NODES, 256, 0, stream>>>(adj, fsrc, fdst, mrow, lrow);
    LML_attwh_kernel<<<dim3(64, 4), 128, 0, stream>>>(adj, fsrc, fdst, mrow, lrow, Wh, Hp4);
    LML_reduce4_kernel<<<(N_NODES * F_OUT + 255) / 256, 256, 0, stream>>>(Hp4, Hp, N_NODES * F_OUT);
    LML_attnout_kernel<<<dim3(N_NODES / 16, 4), 256, 0, stream>>>(Hp, fc_w, fc_b, x, x1);

    // --- MLP branch ---
    LML_ln_kernel<<<N_NODES, 256, 0, stream>>>(x1, ln2_g, ln2_b, hbuf);
    LML_fc1_kernel<<<dim3(N_NODES / 16, 8), 256, 0, stream>>>(hbuf, fc1_w, fc1_b, mbuf);
    LML_fc2_kernel<<<dim3(N_NODES / 16, 4), 256, 0, stream>>>(mbuf, fc2_w, fc2_b, x1, out);
}